// RaySamplePoint_34076270527091
// MI455X (gfx1250) — compile-verified
//
#include <hip/hip_runtime.h>
#include <stdint.h>

#define NRAYS 524288
#define NS    64
#define RPB   8            // rays per block (1 ray per wave32)
#define TPB   256          // 8 waves

typedef float v2f  __attribute__((ext_vector_type(2)));
typedef float v4f  __attribute__((ext_vector_type(4)));
typedef unsigned int u32x4 __attribute__((ext_vector_type(4)));
typedef int   i32x4 __attribute__((ext_vector_type(4)));
typedef int   i32x8 __attribute__((ext_vector_type(8)));
typedef __attribute__((address_space(3))) float lds_float_t;

__global__ __launch_bounds__(TPB)
void RaySamplePoint_kernel(const float* __restrict__ rays,
                           const float* __restrict__ bbox,
                           const float* __restrict__ bins,
                           float* __restrict__ out_t,
                           float* __restrict__ out_p,
                           float* __restrict__ out_m)
{
    __shared__ float s_ray[RPB * 6];                       // ro.xyz, rd.xyz per ray
    __shared__ float s_box[RPB * 6];                       // lo.xyz, hi.xyz per ray
    __shared__ __align__(16) float s_pts[RPB * NS * 3];    // 1536 floats = 6144 B

    const int tid = threadIdx.x;
    const int R   = blockIdx.x * RPB;

    // ---- stage per-block ray + bbox(lo/hi) data into LDS ----
    if (tid < RPB * 6) {
        const int r = tid / 6, k = tid % 6;
        s_ray[tid] = rays[(size_t)(R + r) * 6 + k];
        // corner0 (lo) comps 0..2, corner6 (hi) comps 0..2 -> offsets 0,1,2,18,19,20
        s_box[tid] = bbox[(size_t)(R + r) * 24 + (k < 3 ? k : 15 + k)];
    }
    __syncthreads();

    const int w = tid >> 5;        // wave id == ray within block
    const int l = tid & 31;        // lane
    const int n = R + w;           // global ray id

    const float rox = s_ray[w*6+0], roy = s_ray[w*6+1], roz = s_ray[w*6+2];
    const float rdx = s_ray[w*6+3], rdy = s_ray[w*6+4], rdz = s_ray[w*6+5];
    const float lox = s_box[w*6+0], loy = s_box[w*6+1], loz = s_box[w*6+2];
    const float hix = s_box[w*6+3], hiy = s_box[w*6+4], hiz = s_box[w*6+5];

    const float EPS = 2.220446049250313e-16f;
    const float ix = rdx + EPS, iy = rdy + EPS, iz = rdz + EPS;

    float tt[6];
    tt[0] = (lox - rox) / ix;  tt[1] = (hix - rox) / ix;
    tt[2] = (loy - roy) / iy;  tt[3] = (hiy - roy) / iy;
    tt[4] = (loz - roz) / iz;  tt[5] = (hiz - roz) / iz;

    bool ok[6];
    #pragma unroll
    for (int i = 0; i < 6; ++i) {
        const float px = tt[i] * rdx + rox;
        const float py = tt[i] * rdy + roy;
        const float pz = tt[i] * rdz + roz;
        if (i < 2)       ok[i] = (py >= loy) & (py <= hiy) & (pz >= loz) & (pz <= hiz); // x faces
        else if (i < 4)  ok[i] = (px >= lox) & (px <= hix) & (pz >= loz) & (pz <= hiz); // y faces
        else             ok[i] = (px >= lox) & (px <= hix) & (py >= loy) & (py <= hiy); // z faces
    }

    // top-2 of masked t list (matches jax.lax.top_k incl. duplicates)
    float m1 = -1.0e30f, m2 = -1.0e30f;
    #pragma unroll
    for (int i = 0; i < 6; ++i) {
        const float v = ok[i] ? tt[i] : -1000.0f;
        if (v > m1)      { m2 = m1; m1 = v; }
        else if (v > m2) { m2 = v; }
    }
    const float start = m2;
    const float bw    = (m1 - start) * (1.0f / NS);   // /64 is exact power-of-2

    if (l == 0) out_m[n] = (fabsf(bw) > 1e-5f) ? 1.0f : 0.0f;

    // ---- 2 samples per lane: dense NT b64 load + store ----
    const int s0 = l * 2;
    const v2f bs = __builtin_nontemporal_load((const v2f*)(bins + (size_t)n * NS + s0));
    const float ta = ((float)(s0)     + bs.x) * bw + start;
    const float tb = ((float)(s0 + 1) + bs.y) * bw + start;
    v2f tp; tp.x = ta; tp.y = tb;
    __builtin_nontemporal_store(tp, (v2f*)(out_t + (size_t)n * NS + s0));

    float* sp = &s_pts[(w * NS + s0) * 3];
    sp[0] = ta * rdx + rox;  sp[1] = ta * rdy + roy;  sp[2] = ta * rdz + roz;
    sp[3] = tb * rdx + rox;  sp[4] = tb * rdy + roy;  sp[5] = tb * rdz + roz;

    __syncthreads();   // all point tiles visible in LDS

#if __has_builtin(__builtin_amdgcn_tensor_store_from_lds)
    // ---- TDM: one 1-D descriptor DMAs the block's 1536 contiguous floats ----
    if (tid < 32) {   // wave 0 issues; EXEC-uniform branch
        const uint32_t lds_off = (uint32_t)(size_t)(lds_float_t*)s_pts;
        const uint64_t ga      = (uint64_t)(size_t)(out_p + (size_t)R * NS * 3);
        const uint32_t nelem   = RPB * NS * 3;   // 1536, fits tile_dim0[15:0]

        u32x4 g0;
        g0[0] = 1u;                                   // count=1, user mode
        g0[1] = lds_off;                              // lds_addr
        g0[2] = (uint32_t)ga;                         // global_addr[31:0]
        g0[3] = (uint32_t)((ga >> 32) & 0x1FFFFFFu)   // global_addr[56:32]
              | (2u << 30);                           // type = 2 ("image")

        i32x8 g1;
        g1[0] = (int)(2u << 16);                      // data_size = 4 bytes
        g1[1] = (int)((nelem & 0xFFFFu) << 16);       // tensor_dim0[15:0]
        g1[2] = (int)((nelem >> 16) | (1u << 16));    // tensor_dim0[31:16], tensor_dim1=1
        g1[3] = (int)((nelem & 0xFFFFu) << 16);       // tile_dim0 = 1536
        g1[4] = 0;                                    // tile_dim1/2 unused -> 1-D
        g1[5] = (int)nelem;                           // tensor_dim0_stride
        g1[6] = 0;
        g1[7] = 0;

        i32x4 z4; z4[0] = 0; z4[1] = 0; z4[2] = 0; z4[3] = 0;
        i32x8 z8; z8[0] = 0; z8[1] = 0; z8[2] = 0; z8[3] = 0;
                  z8[4] = 0; z8[5] = 0; z8[6] = 0; z8[7] = 0;
        // clang-23 / therock-10.0 6-arg form:
        // (u32x4 g0, i32x8 g1, i32x4 g2, i32x4 g3, i32x8 extra, i32 cpol)
        __builtin_amdgcn_tensor_store_from_lds(g0, g1, z4, z4, z8, 0);
#if __has_builtin(__builtin_amdgcn_s_wait_tensorcnt)
        __builtin_amdgcn_s_wait_tensorcnt(0);
#endif
    }
#else
    // fallback: cooperative 16B-aligned NT b128 stores straight from LDS
    for (int i = tid; i < RPB * NS * 3 / 4; i += TPB) {
        const v4f v = *(const v4f*)&s_pts[i * 4];
        __builtin_nontemporal_store(v, (v4f*)(out_p + (size_t)R * NS * 3 + i * 4));
    }
#endif
}

extern "C" void kernel_launch(void* const* d_in, const int* in_sizes, int n_in,
                              void* d_out, int out_size, void* d_ws, size_t ws_size,
                              hipStream_t stream) {
    const float* rays = (const float*)d_in[0];   // (N,6)
    const float* bbox = (const float*)d_in[1];   // (N,8,3)
    const float* bins = (const float*)d_in[2];   // (N,64)
    float* out_t = (float*)d_out;                           // N*64
    float* out_p = (float*)d_out + (size_t)NRAYS * NS;      // N*64*3
    float* out_m = out_p + (size_t)NRAYS * NS * 3;          // N
    RaySamplePoint_kernel<<<dim3(NRAYS / RPB), dim3(TPB), 0, stream>>>(
        rays, bbox, bins, out_t, out_p, out_m);
}